// VisualRetina_1468878815912
// MI455X (gfx1250) — compile-verified
//
#include <hip/hip_runtime.h>
#include <math.h>

typedef __attribute__((ext_vector_type(2))) float v2f;
typedef __attribute__((ext_vector_type(8))) float v8f;

#define IMG_H 224
#define IMG_W 224
#define CIN   3
#define NCHUNK 64

// One wave (32 lanes) handles one 16-chunk column tile (tau) of one batch image,
// looping over the 3 channels. 4 waves per block => block == one batch image.
__global__ __launch_bounds__(128) void retina_kernel(const float* __restrict__ x,
                                                     float* __restrict__ out)
{
    const int b    = blockIdx.x;
    const int lane = threadIdx.x & 31;
    const int tau  = threadIdx.x >> 5;          // column tile 0..3

    __shared__ float ylds[4][16 * 16];          // per-wave Y tile, [t][nloc]
    __shared__ float cst [4][4 * 16];           // per-wave coeff staging (rows 0..3)

    // ----- closed-form T_pinv of T = [t^3 t^2 t 1], t = linspace(-1,1,16) -----
    float S2 = 0.f, S4 = 0.f, S6 = 0.f;
    float tv[16];
    #pragma unroll
    for (int j = 0; j < 16; ++j) {
        float t = -1.f + 2.f * (float)j / 15.f;
        tv[j] = t;
        float t2 = t * t;
        S2 += t2; S4 += t2 * t2; S6 += t2 * t2 * t2;
    }
    const float D1 = S6 * S2 - S4 * S4;         // couples {t^3, t}
    const float D2 = 16.f * S4 - S2 * S2;       // couples {t^2, 1}
    auto tpinv = [&](int k, int j) -> float {   // T_pinv[k][j]
        float t = tv[j], t2 = t * t, t3 = t2 * t;
        switch (k) {
            case 0:  return ( S2   * t3 - S4 * t) / D1;   // t^3 coeff
            case 1:  return ( 16.f * t2 - S2    ) / D2;   // t^2 coeff
            case 2:  return (-S4   * t3 + S6 * t) / D1;   // t   coeff
            default: return (-S2   * t2 + S4    ) / D2;   // const coeff
        }
    };

    const int  mrow = lane & 15;
    const bool lo   = lane < 16;

    // A fragments for coeff GEMM: A_j = Tpinv_pad[:, 4j:4j+4] (16x4, rows>=4 zero)
    // f32 A 16x4 layout: v0 = K0 | K2 (upper lanes), v1 = K1 | K3
    v2f afrag[4];
    #pragma unroll
    for (int j = 0; j < 4; ++j) {
        float a0 = 0.f, a1 = 0.f;
        if (mrow < 4) {
            a0 = tpinv(mrow, 4 * j + (lo ? 0 : 2));
            a1 = tpinv(mrow, 4 * j + (lo ? 1 : 3));
        }
        afrag[j][0] = a0; afrag[j][1] = a1;
    }
    // A fragment for y_pred GEMM: T_mat (16x4), T[t][k] = t^(3-k)
    v2f tafrag;
    {
        float t = tv[mrow], t2 = t * t;
        tafrag[0] = lo ? t2 * t : t;
        tafrag[1] = lo ? t2     : 1.f;
    }

    float sum_c0 = 0.f;   // lanes<16: coeff k0 ; lanes>=16: coeff k2  (n = 16*tau+mrow)
    float sum_c1 = 0.f;   // lanes<16: coeff k1 ; lanes>=16: coeff k3
    float sum_sig = 0.f;  // sigma (valid everywhere, written from lanes<16)

    for (int c = 0; c < CIN; ++c) {
        __syncthreads();  // previous iteration's LDS readers are done

        // ---- gather 256 samples: z-order positions p = 256*tau .. +255 ------
        const size_t chanBase = (size_t)(b * CIN + c) * (IMG_H * IMG_W);
        #pragma unroll
        for (int i = 0; i < 8; ++i) {
            int p  = tau * 256 + i * 32 + lane;
            int xg = 0, yg = 0;                 // Morton decode: x=even bits, y=odd
            #pragma unroll
            for (int k = 0; k < 5; ++k) {
                xg |= ((p >> (2 * k    )) & 1) << k;
                yg |= ((p >> (2 * k + 1)) & 1) << k;
            }
            float v = x[chanBase + (size_t)(7 * yg) * IMG_W + (size_t)(7 * xg)];
            ylds[tau][(p & 15) * 16 + ((p >> 4) & 15)] = v;   // [t][nloc]
        }
        __syncthreads();

        // ---- B fragments (Y rows, K-slices) and C fragment (-Y) -------------
        v2f bfrag[4];
        #pragma unroll
        for (int j = 0; j < 4; ++j) {
            int k0 = 4 * j + (lo ? 0 : 2);
            bfrag[j][0] = ylds[tau][ k0      * 16 + mrow];
            bfrag[j][1] = ylds[tau][(k0 + 1) * 16 + mrow];
        }
        v8f cneg;
        #pragma unroll
        for (int r = 0; r < 8; ++r)
            cneg[r] = -ylds[tau][(lo ? r : r + 8) * 16 + mrow];

        // ---- coeffs = T_pinv @ Y : four chained K=4 f32 WMMAs ---------------
        v8f acc = {};
        #pragma unroll
        for (int j = 0; j < 4; ++j) {
            acc = __builtin_amdgcn_wmma_f32_16x16x4_f32(
                false, afrag[j], false, bfrag[j], (short)0, acc, false, false);
        }

        // stage coeff rows 0..3 (lanes 0..15 of acc[0..3]) -> B-layout reload
        if (lo) {
            cst[tau][0 * 16 + mrow] = acc[0];
            cst[tau][1 * 16 + mrow] = acc[1];
            cst[tau][2 * 16 + mrow] = acc[2];
            cst[tau][3 * 16 + mrow] = acc[3];
        }
        __syncthreads();

        v2f bc;
        bc[0] = cst[tau][(lo ? 0 : 2) * 16 + mrow];
        bc[1] = cst[tau][(lo ? 1 : 3) * 16 + mrow];
        sum_c0 += bc[0];
        sum_c1 += bc[1];

        // ---- (y_pred - y) = T_mat @ coeffs + (-Y), fused in accumulator -----
        v8f d = __builtin_amdgcn_wmma_f32_16x16x4_f32(
            false, tafrag, false, bc, (short)0, cneg, false, false);

        float ss = 0.f;
        #pragma unroll
        for (int r = 0; r < 8; ++r) ss += d[r] * d[r];
        ss += __shfl_xor(ss, 16, 32);           // fold t=0..7 with t=8..15 halves
        sum_sig += sqrtf(ss * (1.0f / 16.0f));
    }

    // ---- output: (B, 64, 1, 5) = [c0, c1, c2, c3, sigma], channel-mean ------
    const int n = tau * 16 + mrow;
    float* o = out + ((size_t)b * NCHUNK + n) * 5;
    const float inv3 = 1.0f / 3.0f;
    if (lo) {
        o[0] = sum_c0 * inv3;
        o[1] = sum_c1 * inv3;
        o[4] = sum_sig * inv3;
    } else {
        o[2] = sum_c0 * inv3;
        o[3] = sum_c1 * inv3;
    }
}

extern "C" void kernel_launch(void* const* d_in, const int* in_sizes, int n_in,
                              void* d_out, int out_size, void* d_ws, size_t ws_size,
                              hipStream_t stream) {
    (void)in_sizes; (void)n_in; (void)d_ws; (void)ws_size; (void)out_size;
    const float* x = (const float*)d_in[0];
    float* out = (float*)d_out;
    retina_kernel<<<256, 128, 0, stream>>>(x, out);
}